// RetinaLoss_37306085933328
// MI455X (gfx1250) — compile-verified
//
#include <hip/hip_runtime.h>
#include <cstdint>
#include <cstddef>

// ---------------- constants from the reference ----------------
#define ALPHA_F   0.25f
#define BETA_F    (1.0f / 9.0f)
#define EPS_F     1.0e-4f
#define IMG_WH    640.0f
#define C_CLS     80
#define M_GT      16
#define BLOCK     128
#define ROWPAD    84   // dwords per LDS row: 80 data + 4 pad (336B, 16B aligned, 2-way bank conflicts max)

// ---------------- CDNA5 async global->LDS helpers (inline asm, gfx1250) ----------------
// GLOBAL_LOAD_ASYNC_TO_LDS_B128 (ISA 15.18.3 opcode 98): vdst = LDS byte address VGPR,
// vaddr = 64-bit global address. INST_OFFSET is added to BOTH the LDS and the global
// address (ISA 08 §4.4), so one base VGPR (each side) + per-chunk immediate covers a row.
template <int K>
struct AsyncRowCopy {
    static __device__ __forceinline__ void run(uint32_t lds_base, uint64_t g_base) {
        AsyncRowCopy<K - 1>::run(lds_base, g_base);
        asm volatile("global_load_async_to_lds_b128 %0, %1, off offset:%2"
                     :
                     : "v"(lds_base), "v"(g_base), "i"((K - 1) * 16)
                     : "memory");
    }
};
template <>
struct AsyncRowCopy<0> {
    static __device__ __forceinline__ void run(uint32_t, uint64_t) {}
};

__device__ __forceinline__ void wait_async_zero() {
    asm volatile("s_wait_asynccnt 0" ::: "memory");
}

// ---------------- main fused kernel ----------------
__global__ __launch_bounds__(BLOCK)
void retina_main(const float* __restrict__ cls_heads,   // (B,N,80)
                 const float* __restrict__ reg_heads,   // (B,N,4)
                 const float* __restrict__ anchors,     // (B,N,4)
                 const float* __restrict__ annots,      // (B,16,5)
                 int N, int B,
                 int*   __restrict__ pos_cnt,
                 float* __restrict__ cls_sum,
                 float* __restrict__ reg_sum)
{
    __shared__ float s_cls[BLOCK * ROWPAD];   // 43008 B
    __shared__ float s_ann[M_GT * 5];

    const int  tid    = threadIdx.x;
    const int  b      = blockIdx.y;
    const int  n      = blockIdx.x * BLOCK + tid;
    const bool active = (n < N);

    // ---- stage this block's cls tile into LDS via async DMA (issued first, earliest start) ----
    if (active) {
        const uint64_t grow = (uint64_t)(uintptr_t)(cls_heads + ((size_t)b * N + n) * C_CLS);
        const uint32_t lrow = (uint32_t)(uintptr_t)(&s_cls[tid * ROWPAD]);
        AsyncRowCopy<C_CLS / 4>::run(lrow, grow);   // 20 x b128, offsets 0..304 as immediates
    }
    // ---- stage annotations (16 x 5 floats) ----
    if (tid < M_GT * 5) s_ann[tid] = annots[(size_t)b * M_GT * 5 + tid];

    wait_async_zero();     // this wave's async copies into LDS are done
    __syncthreads();       // all waves' LDS writes visible

    float  label  = -1.0f;
    float4 tgt    = make_float4(0.f, 0.f, 0.f, 0.f);
    float  fl_sum = 0.0f;
    float  rl_sum = 0.0f;

    if (active) {
        // ------------- anchor / GT assignment -------------
        const float4 a = ((const float4*)anchors)[(size_t)b * N + n];
        const float aw = a.z - a.x, ah = a.w - a.y;
        const float area_a = aw * ah;

        float best = -1.0f; int bi = 0; bool any_valid = false;
#pragma unroll
        for (int m = 0; m < M_GT; ++m) {
            const float gx0 = s_ann[m * 5 + 0], gy0 = s_ann[m * 5 + 1];
            const float gx1 = s_ann[m * 5 + 2], gy1 = s_ann[m * 5 + 3];
            const float gc  = s_ann[m * 5 + 4];
            const bool  valid = (gc >= 0.0f);
            any_valid = any_valid || valid;
            float iw = fmaxf(fminf(a.z, gx1) - fmaxf(a.x, gx0), 0.0f);
            float ih = fmaxf(fminf(a.w, gy1) - fmaxf(a.y, gy0), 0.0f);
            float inter  = iw * ih;
            float area_g = (gx1 - gx0) * (gy1 - gy0);
            float iou    = inter / (area_a + area_g - inter);
            iou = valid ? iou : -1.0f;
            if (iou > best) { best = iou; bi = m; }   // strict '>' keeps first argmax
        }

        const float gx0 = s_ann[bi * 5 + 0], gy0 = s_ann[bi * 5 + 1];
        const float gx1 = s_ann[bi * 5 + 2], gy1 = s_ann[bi * 5 + 3];
        const float gcls = s_ann[bi * 5 + 4];
        const float acx = a.x + 0.5f * aw, acy = a.y + 0.5f * ah;
        const float gw  = fmaxf(gx1 - gx0, 1.0f);
        const float gh  = fmaxf(gy1 - gy0, 1.0f);
        const float gcx = gx0 + 0.5f * gw, gcy = gy0 + 0.5f * gh;
        tgt.x = ((gcx - acx) / aw) * 10.0f;       // / 0.1
        tgt.y = ((gcy - acy) / ah) * 10.0f;       // / 0.1
        tgt.z = __logf(gw / aw) * 5.0f;           // / 0.2
        tgt.w = __logf(gh / ah) * 5.0f;           // / 0.2

        label = -1.0f;
        if (best < 0.4f) label = 0.0f;
        if (best > 0.5f) label = gcls + 1.0f;
        if (!any_valid)  label = -1.0f;
        const bool inside = (a.x > 0.0f) && (a.y > 0.0f) && (a.z < IMG_WH) && (a.w < IMG_WH);
        if (!inside) label = -1.0f;

        // ------------- focal loss over 80 classes (from LDS) -------------
        if (label >= 0.0f) {
            const int     tcls = (int)label - 1;   // -1 when label==0 (background)
            const float4* row  = (const float4*)(&s_cls[tid * ROWPAD]);
#pragma unroll
            for (int k = 0; k < C_CLS / 4; ++k) {
                const float4 v = row[k];
                const float pv[4] = { v.x, v.y, v.z, v.w };
#pragma unroll
                for (int j = 0; j < 4; ++j) {
                    const int   c  = k * 4 + j;
                    const float p  = fminf(fmaxf(pv[j], EPS_F), 1.0f - EPS_F);
                    const bool  gp = (c == tcls);
                    const float pt = gp ? p : (1.0f - p);
                    const float af = gp ? ALPHA_F : (1.0f - ALPHA_F);
                    const float om = 1.0f - pt;
                    fl_sum += af * om * om * (-__logf(pt));
                }
            }
        }

        // ------------- smooth-L1 on positives -------------
        if (label > 0.0f) {
            const float4 r = ((const float4*)reg_heads)[(size_t)b * N + n];
            const float dx[4] = { fabsf(r.x - tgt.x), fabsf(r.y - tgt.y),
                                  fabsf(r.z - tgt.z), fabsf(r.w - tgt.w) };
#pragma unroll
            for (int j = 0; j < 4; ++j) {
                const float x = dx[j];
                rl_sum += (x < BETA_F) ? (0.5f * x * x / BETA_F) : (x - 0.5f * BETA_F);
            }
        }
    }

    // ------------- wave32 reductions + one atomic per wave -------------
    const bool pos_anchor = active && (label > 0.0f);
    const unsigned long long bal = __ballot(pos_anchor);
    float fw = fl_sum, rw = rl_sum;
#pragma unroll
    for (int o = 16; o > 0; o >>= 1) {
        fw += __shfl_xor(fw, o, 32);
        rw += __shfl_xor(rw, o, 32);
    }
    if ((tid & 31) == 0) {
        if (fw != 0.0f) atomicAdd(&cls_sum[b], fw);
        if (rw != 0.0f) atomicAdd(&reg_sum[b], rw);
        const int pc = (int)__popcll(bal);
        if (pc) atomicAdd(&pos_cnt[b], pc);
    }
}

// ---------------- small helper kernels ----------------
__global__ void retina_init(int* __restrict__ pos_cnt, float* __restrict__ cls_sum,
                            float* __restrict__ reg_sum, int B) {
    int i = threadIdx.x;
    if (i < B) { pos_cnt[i] = 0; cls_sum[i] = 0.0f; reg_sum[i] = 0.0f; }
}

__global__ void retina_final(const int* __restrict__ pos_cnt,
                             const float* __restrict__ cls_sum,
                             const float* __restrict__ reg_sum,
                             int B, float* __restrict__ out)
{
    if (threadIdx.x == 0 && blockIdx.x == 0) {
        float ca = 0.0f, ra = 0.0f, nv = 0.0f;
        for (int b = 0; b < B; ++b) {
            const float p = (float)pos_cnt[b];
            if (p > 0.0f) {
                ca += cls_sum[b] / fmaxf(p, 1.0f);
                ra += reg_sum[b] / fmaxf(4.0f * p, 1.0f);
                nv += 1.0f;
            }
        }
        nv = fmaxf(nv, 1.0f);
        out[0] = ca / nv;
        out[1] = ra / nv;
    }
}

// ---------------- host entry ----------------
extern "C" void kernel_launch(void* const* d_in, const int* in_sizes, int n_in,
                              void* d_out, int out_size, void* d_ws, size_t ws_size,
                              hipStream_t stream)
{
    const float* cls_heads = (const float*)d_in[0];   // (B,N,80)
    const float* reg_heads = (const float*)d_in[1];   // (B,N,4)
    const float* anchors   = (const float*)d_in[2];   // (B,N,4)
    const float* annots    = (const float*)d_in[3];   // (B,16,5)

    const int B = in_sizes[3] / (M_GT * 5);           // = 4
    const int N = in_sizes[1] / (B * 4);              // = 76725

    // workspace layout: [pos_cnt: B ints @0] [cls_sum: B floats @256] [reg_sum: B floats @512]
    int*   pos_cnt = (int*)d_ws;
    float* cls_sum = (float*)((char*)d_ws + 256);
    float* reg_sum = (float*)((char*)d_ws + 512);

    retina_init<<<1, 32, 0, stream>>>(pos_cnt, cls_sum, reg_sum, B);

    dim3 grid((N + BLOCK - 1) / BLOCK, B, 1);
    retina_main<<<grid, BLOCK, 0, stream>>>(cls_heads, reg_heads, anchors, annots,
                                            N, B, pos_cnt, cls_sum, reg_sum);

    retina_final<<<1, 32, 0, stream>>>(pos_cnt, cls_sum, reg_sum, B, (float*)d_out);
}